// ConditionalLayers_68839735820624
// MI455X (gfx1250) — compile-verified
//
#include <hip/hip_runtime.h>

typedef float v2f __attribute__((ext_vector_type(2)));
typedef float v8f __attribute__((ext_vector_type(8)));

#define NUM_COND 32
#define CIN      64
#define COUT     64
#define BDIM     128
#define TDIM     8192
#define T_BLK    128
// Interleaved pair-row layout: W[c,d] -> sW[(c>>1)*PAIR_STRIDE + 2*d + (c&1)]
// PAIR_STRIDE = 2*64 + 32 so consecutive pair-rows shift banks by 32:
// half-wave 0 (rows c,c+1) and half-wave 1 (rows c+2,c+3) hit disjoint bank sets.
#define PAIR_STRIDE 160

__global__ __launch_bounds__(256, 1)
void ConditionalLayers_68839735820624_kernel(const float* __restrict__ x,
                                             const float* __restrict__ w,
                                             const int*   __restrict__ cond,
                                             float*       __restrict__ out)
{
    __shared__ float sW[(CIN / 2) * PAIR_STRIDE];   // 20 KB

    const int b   = blockIdx.y;
    const int t0  = blockIdx.x * T_BLK;
    const int tid = threadIdx.x;

    // Gather this sample's weight matrix [CIN, COUT] into LDS, K-pair interleaved.
    const int cid = cond[b];
    const float* __restrict__ wg = w + (size_t)cid * (CIN * COUT);
    #pragma unroll
    for (int i = 0; i < 4; ++i) {
        const int j  = tid + i * 256;        // float4 index, 0..1023
        const int c  = j >> 4;               // 16 float4 per 64-float row
        const int d4 = (j & 15) << 2;
        const float4 v = ((const float4*)wg)[j];
        float* dst = &sW[(c >> 1) * PAIR_STRIDE + (d4 << 1) + (c & 1)];
        dst[0] = v.x; dst[2] = v.y; dst[4] = v.z; dst[6] = v.w;
    }
    __syncthreads();

    const int wv   = tid >> 5;     // wave id 0..7 -> 16-wide t-tile
    const int lane = tid & 31;
    const int half = lane >> 4;    // K half: rows (k0+2h, k0+2h+1)
    const int lt   = lane & 15;    // M (for A) / N (for B,D) position

    const int tcol = t0 + wv * 16 + lt;
    const float* __restrict__ xb = x + (size_t)b * CIN * TDIM + tcol;

    v8f acc0 = {}, acc1 = {}, acc2 = {}, acc3 = {};

    #pragma unroll
    for (int k0 = 0; k0 < CIN; k0 += 4) {
        const int c = k0 + 2 * half;         // even

        // B fragment (4x16 f32): lane holds column t, K = c, c+1 (streamed, NT)
        v2f bf;
        bf.x = __builtin_nontemporal_load(&xb[(size_t)c       * TDIM]);
        bf.y = __builtin_nontemporal_load(&xb[(size_t)(c + 1) * TDIM]);

        // A fragments: {W[c,d], W[c+1,d]} are adjacent in LDS -> one ds_load_b64 each
        const float* pr = &sW[(c >> 1) * PAIR_STRIDE + (lt << 1)];
        const v2f a0 = *(const v2f*)(pr);
        const v2f a1 = *(const v2f*)(pr + 32);
        const v2f a2 = *(const v2f*)(pr + 64);
        const v2f a3 = *(const v2f*)(pr + 96);

        acc0 = __builtin_amdgcn_wmma_f32_16x16x4_f32(false, a0, false, bf, (short)0, acc0, false, false);
        acc1 = __builtin_amdgcn_wmma_f32_16x16x4_f32(false, a1, false, bf, (short)0, acc1, false, false);
        acc2 = __builtin_amdgcn_wmma_f32_16x16x4_f32(false, a2, false, bf, (short)0, acc2, false, false);
        acc3 = __builtin_amdgcn_wmma_f32_16x16x4_f32(false, a3, false, bf, (short)0, acc3, false, false);
    }

    // D layout: VGPR r, lanes 0-15 -> M=r, lanes 16-31 -> M=r+8; N = lt. Streaming NT stores.
    float* __restrict__ ob = out + (size_t)b * COUT * TDIM + tcol;
    #pragma unroll
    for (int r = 0; r < 8; ++r) {
        const int d = r + 8 * half;
        __builtin_nontemporal_store(acc0[r], &ob[(size_t)(d)      * TDIM]);
        __builtin_nontemporal_store(acc1[r], &ob[(size_t)(d + 16) * TDIM]);
        __builtin_nontemporal_store(acc2[r], &ob[(size_t)(d + 32) * TDIM]);
        __builtin_nontemporal_store(acc3[r], &ob[(size_t)(d + 48) * TDIM]);
    }
}

extern "C" void kernel_launch(void* const* d_in, const int* in_sizes, int n_in,
                              void* d_out, int out_size, void* d_ws, size_t ws_size,
                              hipStream_t stream) {
    const float* x    = (const float*)d_in[0];
    const float* wts  = (const float*)d_in[1];
    const int*   cids = (const int*)d_in[2];
    float*       out  = (float*)d_out;

    dim3 grid(TDIM / T_BLK, BDIM);   // 64 x 128 = 8192 blocks
    ConditionalLayers_68839735820624_kernel<<<grid, 256, 0, stream>>>(x, wts, cids, out);
}